// RandScatter_16716012716274
// MI455X (gfx1250) — compile-verified
//
#include <hip/hip_runtime.h>

// Top-1 MoE capacity dispatch: N=16384 tokens, D=1024 f32, P=8 paths, CAP=4096.
// Memory-bound (~192 MB total traffic -> ~8.3us @ 23.3 TB/s). Strategy:
//   1) per-token argmax + global per-path rank (wave32 ballot + block hist + scan)
//   2) invert to slot[] = source token per output row (gather formulation:
//      128 MB written once, vs 192 MB for zero-fill + scatter)
//   3) bulk row copy via CDNA5 async global<->LDS DMA (ASYNCcnt).

#define N_TOK 16384
#define D_DIM 1024
#define P_NUM 8
#define CAP   4096
#define BLK   256
#define NBLK  (N_TOK / BLK)   // 64

#if defined(__HIP_DEVICE_COMPILE__)
#if __has_builtin(__builtin_amdgcn_global_load_async_to_lds_b128) && \
    __has_builtin(__builtin_amdgcn_global_store_async_from_lds_b128)
#define USE_ASYNC_LDS 1
#endif
#endif

__device__ __forceinline__ unsigned ballot32(bool b) {
#if defined(__HIP_DEVICE_COMPILE__) && __has_builtin(__builtin_amdgcn_ballot_w32)
    return __builtin_amdgcn_ballot_w32(b);
#else
    return (unsigned)__ballot((int)b);
#endif
}

// ---------------------------------------------------------------------------
// Kernel 1: per-token path (argmax, first-max tie-break) and global-order rank
// within that path, via wave32 ballots + per-block LDS counters. Also emits a
// per-block 8-bin histogram for the global scan.
// ---------------------------------------------------------------------------
__global__ void __launch_bounds__(BLK)
k_pathrank(const float* __restrict__ score, int* __restrict__ path,
           int* __restrict__ rank, int* __restrict__ hist) {
    __shared__ int cnt[BLK / 32][P_NUM];   // [wave][path]
    const int tid = threadIdx.x;
    const int i   = blockIdx.x * BLK + tid;

    const float* s = score + (size_t)i * P_NUM;
    float best = s[0];
    int   p    = 0;
#pragma unroll
    for (int j = 1; j < P_NUM; ++j) {
        float v = s[j];
        if (v > best) { best = v; p = j; }   // strict '>' == first occurrence
    }

    const int      lane  = tid & 31;
    const int      wave  = tid >> 5;
    const unsigned below = (1u << lane) - 1u;

    unsigned mymask = 0u;
#pragma unroll
    for (int q = 0; q < P_NUM; ++q) {
        unsigned m = ballot32(p == q);
        if (p == q)    mymask = m;
        if (lane == q) cnt[wave][q] = __builtin_popcount(m);
    }
    __syncthreads();

    int r = __builtin_popcount(mymask & below);
    for (int w = 0; w < wave; ++w) r += cnt[w][p];

    path[i] = p;
    rank[i] = r;   // intra-block rank; global offset added later

    if (tid < P_NUM) {
        int t = 0;
#pragma unroll
        for (int w = 0; w < BLK / 32; ++w) t += cnt[w][tid];
        hist[blockIdx.x * P_NUM + tid] = t;
    }
}

// ---------------------------------------------------------------------------
// Kernel 2 (single block): exclusive scan of 64 block-histograms per path,
// and re-init slot[] = -1 (fresh every call -> deterministic under replay).
// ---------------------------------------------------------------------------
__global__ void __launch_bounds__(BLK)
k_scan_init(const int* __restrict__ hist, int* __restrict__ boff,
            int* __restrict__ slot) {
    const int tid = threadIdx.x;
    if (tid < P_NUM) {
        int run = 0;
        for (int b = 0; b < NBLK; ++b) {
            boff[b * P_NUM + tid] = run;
            run += hist[b * P_NUM + tid];
        }
    }
    for (int s = tid; s < P_NUM * CAP; s += BLK) slot[s] = -1;
}

// ---------------------------------------------------------------------------
// Kernel 3: invert the mapping: slot[path*CAP + pos] = token index.
// Positions are unique per path, so no write conflicts.
// ---------------------------------------------------------------------------
__global__ void __launch_bounds__(BLK)
k_scatter(const int* __restrict__ path, const int* __restrict__ rank,
          const int* __restrict__ boff, int* __restrict__ slot) {
    const int i   = blockIdx.x * BLK + threadIdx.x;
    const int p   = path[i];
    const int pos = rank[i] + boff[blockIdx.x * P_NUM + p];
    if (pos < CAP) slot[p * CAP + pos] = i;
}

// ---------------------------------------------------------------------------
// Kernel 4: one block per output row (P*CAP = 32768 rows of 4 KB).
// slot[blockIdx] is a uniform scalar load. Row copy uses the CDNA5 async
// global->LDS->global DMA path (ASYNCcnt); each lane moves its own 16 B so a
// per-wave s_wait_asynccnt suffices (no barrier). Empty rows store zeros
// directly (covers the whole 128 MB output exactly once).
// ---------------------------------------------------------------------------
__global__ void __launch_bounds__(BLK)
k_gather(const float* __restrict__ x, const int* __restrict__ slot,
         float* __restrict__ out) {
    const int s   = blockIdx.x;
    const int tid = threadIdx.x;
    const int t   = slot[s];
    float* dst = out + (size_t)s * D_DIM;

    if (t >= 0) {
        const float* src = x + (size_t)t * D_DIM;
#ifdef USE_ASYNC_LDS
        __shared__ alignas(16) float buf[D_DIM];   // 4 KB row staging
        typedef int v4i __attribute__((vector_size(16)));
        typedef __attribute__((address_space(1))) v4i gv4i;   // global int4
        typedef __attribute__((address_space(3))) v4i lv4i;   // LDS int4
        gv4i* g = (gv4i*)(uintptr_t)(src + tid * 4);
        lv4i* l = (lv4i*)(uintptr_t)(buf + tid * 4);   // low 32 bits = LDS offset
        __builtin_amdgcn_global_load_async_to_lds_b128(g, l, 0, 0);
        asm volatile("s_wait_asynccnt 0" ::: "memory");
        gv4i* gd = (gv4i*)(uintptr_t)(dst + tid * 4);
        __builtin_amdgcn_global_store_async_from_lds_b128(gd, l, 0, 0);
        // outstanding async stores drained by s_endpgm's implicit wait-idle
#else
        const float4* s4 = (const float4*)src;
        float4*       d4 = (float4*)dst;
        d4[tid] = s4[tid];
#endif
    } else {
        ((float4*)dst)[tid] = make_float4(0.f, 0.f, 0.f, 0.f);
    }
}

extern "C" void kernel_launch(void* const* d_in, const int* in_sizes, int n_in,
                              void* d_out, int out_size, void* d_ws, size_t ws_size,
                              hipStream_t stream) {
    const float* x     = (const float*)d_in[0];   // [N, D] f32
    const float* score = (const float*)d_in[1];   // [N, P] f32
    float*       out   = (float*)d_out;           // [P, CAP, D] f32

    int* ws   = (int*)d_ws;
    int* path = ws;                                // N
    int* rank = path + N_TOK;                      // N
    int* hist = rank + N_TOK;                      // NBLK*P
    int* boff = hist + NBLK * P_NUM;               // NBLK*P
    int* slot = boff + NBLK * P_NUM;               // P*CAP

    k_pathrank<<<NBLK, BLK, 0, stream>>>(score, path, rank, hist);
    k_scan_init<<<1, BLK, 0, stream>>>(hist, boff, slot);
    k_scatter<<<NBLK, BLK, 0, stream>>>(path, rank, boff, slot);
    k_gather<<<P_NUM * CAP, BLK, 0, stream>>>(x, slot, out);
}